// PulseRetrievalLossFunction_5995774345558
// MI455X (gfx1250) — compile-verified
//
#include <hip/hip_runtime.h>

// ---- problem constants (from reference setup) ----
constexpr int   BROWS = 256;
constexpr int   NCOLS = 65536;
constexpr int   HALF  = NCOLS / 2;      // 32768
constexpr float THRESH = 0.01f;
constexpr float PEN    = 2.0f;

typedef __attribute__((ext_vector_type(2))) float v2f;
typedef __attribute__((ext_vector_type(4))) float v4f;
typedef __attribute__((ext_vector_type(8))) float v8f;

// ---------------------------------------------------------------------------
// Kernel 1: per-row first/last significant index for |label_real|, |label_imag|
// One workgroup (256 threads = 8 wave32) per row; b128 streaming loads.
// Label loads use default RT hint so the 64 MB array lands in the 192 MB L2
// and is re-hit by loss_kernel.
// ---------------------------------------------------------------------------
__global__ void __launch_bounds__(256)
bounds_kernel(const float* __restrict__ label,
              int* __restrict__ fr, int* __restrict__ lr,
              int* __restrict__ fi, int* __restrict__ li)
{
    const int row = blockIdx.x;
    const int t   = threadIdx.x;

    const v4f* lre = (const v4f*)(label + (size_t)row * NCOLS);
    const v4f* lim = (const v4f*)(label + (size_t)row * NCOLS + HALF);

    int minR = HALF, maxR = -1;
    int minI = HALF, maxI = -1;

    for (int i = t; i < HALF / 4; i += 256) {
        const v4f r = lre[i];
        const v4f m = lim[i];
        const int base = 4 * i;

        if (fabsf(r.x) > THRESH) { minR = min(minR, base + 0); maxR = max(maxR, base + 0); }
        if (fabsf(r.y) > THRESH) { minR = min(minR, base + 1); maxR = max(maxR, base + 1); }
        if (fabsf(r.z) > THRESH) { minR = min(minR, base + 2); maxR = max(maxR, base + 2); }
        if (fabsf(r.w) > THRESH) { minR = min(minR, base + 3); maxR = max(maxR, base + 3); }

        if (fabsf(m.x) > THRESH) { minI = min(minI, base + 0); maxI = max(maxI, base + 0); }
        if (fabsf(m.y) > THRESH) { minI = min(minI, base + 1); maxI = max(maxI, base + 1); }
        if (fabsf(m.z) > THRESH) { minI = min(minI, base + 2); maxI = max(maxI, base + 2); }
        if (fabsf(m.w) > THRESH) { minI = min(minI, base + 3); maxI = max(maxI, base + 3); }
    }

    __shared__ int s_minR, s_maxR, s_minI, s_maxI;
    if (t == 0) { s_minR = HALF; s_maxR = -1; s_minI = HALF; s_maxI = -1; }
    __syncthreads();
    atomicMin(&s_minR, minR); atomicMax(&s_maxR, maxR);
    atomicMin(&s_minI, minI); atomicMax(&s_maxI, maxI);
    __syncthreads();

    if (t == 0) {
        // reference semantics: if no element passes, first=0, last=half-1
        const bool anyR = (s_maxR >= 0);
        const bool anyI = (s_maxI >= 0);
        fr[row] = anyR ? s_minR : 0;
        lr[row] = anyR ? s_maxR : (HALF - 1);
        fi[row] = anyI ? s_minI : 0;
        li[row] = anyI ? s_maxI : (HALF - 1);
    }
}

// ---------------------------------------------------------------------------
// Kernel 2: fused weighted MSE (real/imag) + 50 * intensity MSE, per-row mean.
// Phase term is multiplied by 0.0 in the reference -> dropped.
// prediction is read exactly once -> non-temporal loads (NT hint) so the
// one-shot 64 MB stream does not evict the L2-resident label array.
// ---------------------------------------------------------------------------
__global__ void __launch_bounds__(256)
loss_kernel(const float* __restrict__ pred, const float* __restrict__ label,
            const int* __restrict__ fr, const int* __restrict__ lr,
            const int* __restrict__ fi, const int* __restrict__ li,
            float* __restrict__ row_mean)
{
    const int row = blockIdx.x;
    const int t   = threadIdx.x;

    const int frv = fr[row], lrv = lr[row];
    const int fiv = fi[row], liv = li[row];

    const v4f* pre = (const v4f*)(pred  + (size_t)row * NCOLS);
    const v4f* pim = (const v4f*)(pred  + (size_t)row * NCOLS + HALF);
    const v4f* lre = (const v4f*)(label + (size_t)row * NCOLS);
    const v4f* lim = (const v4f*)(label + (size_t)row * NCOLS + HALF);

    float acc = 0.0f;

    for (int i = t; i < HALF / 4; i += 256) {
        const v4f pr = __builtin_nontemporal_load(&pre[i]);   // NT: single-use stream
        const v4f pi = __builtin_nontemporal_load(&pim[i]);   // NT: single-use stream
        const v4f la = lre[i];                                // RT: L2 re-hit
        const v4f lb = lim[i];                                // RT: L2 re-hit
        const int base = 4 * i;

        #define ELEM(c, k)                                                     \
        {                                                                      \
            const int   idx  = base + (k);                                     \
            const float dr   = pr.c - la.c;                                    \
            const float di   = pi.c - lb.c;                                    \
            const float wre  = (idx < frv || idx > lrv) ? PEN : 1.0f;          \
            const float wim  = (idx < fiv || idx > liv) ? PEN : 1.0f;          \
            const float pint = pr.c * pr.c + pi.c * pi.c;                      \
            const float lint = la.c * la.c + lb.c * lb.c;                      \
            const float dint = pint - lint;                                    \
            acc = fmaf(wre * dr, dr, acc);                                     \
            acc = fmaf(wim * di, di, acc);                                     \
            acc = fmaf(50.0f * dint, dint, acc);                               \
        }
        ELEM(x, 0) ELEM(y, 1) ELEM(z, 2) ELEM(w, 3)
        #undef ELEM
    }

    // intra-wave reduction (wave32)
    for (int off = 16; off >= 1; off >>= 1)
        acc += __shfl_down(acc, off, 32);

    __shared__ float s_part[8];
    const int wave = t >> 5;
    const int lane = t & 31;
    if (lane == 0) s_part[wave] = acc;
    __syncthreads();

    if (wave == 0) {
        float v = (lane < 8) ? s_part[lane] : 0.0f;
        for (int off = 4; off >= 1; off >>= 1)
            v += __shfl_down(v, off, 32);
        if (lane == 0)
            row_mean[row] = v * (1.0f / (float)HALF);   // per-sample value
    }
}

// ---------------------------------------------------------------------------
// Kernel 3: reduce the 256 per-row means with V_WMMA_F32_16X16X4_F32.
// A = 16x4 all-ones, B = 4x16 chunk of data, D = 1*B + C accumulates column
// sums. Since A is all ones and we then sum every element of D's first C/D
// row, the result is invariant to the exact K<->lane mapping of B.
// Launched as a single wave32 (EXEC all ones, as WMMA requires).
// ---------------------------------------------------------------------------
__global__ void __launch_bounds__(32)
final_kernel(const float* __restrict__ row_mean, float* __restrict__ out)
{
    const int lane = threadIdx.x;   // 0..31

    v2f a; a[0] = 1.0f; a[1] = 1.0f;     // all-ones A operand (16x4 f32)
    v8f c = {};                          // f32 accumulator (16x16)

    #pragma unroll
    for (int chunk = 0; chunk < 4; ++chunk) {
        v2f b;
        b[0] = row_mean[chunk * 64 + lane];
        b[1] = row_mean[chunk * 64 + 32 + lane];
        // 8 args: (neg_a, A, neg_b, B, c_mod, C, reuse_a, reuse_b)
        c = __builtin_amdgcn_wmma_f32_16x16x4_f32(
                false, a, false, b, (short)0, c, false, false);
    }

    // VGPR0 of C/D: lanes 0-15 hold row M=0 (N=lane), lanes 16-31 hold row
    // M=8 -- identical column sums. Reducing all 32 lanes double-counts, so
    // scale by 0.5. Then divide by BROWS for the final mean.
    float v = c[0];
    for (int off = 16; off >= 1; off >>= 1)
        v += __shfl_xor(v, off, 32);

    if (lane == 0)
        out[0] = v * (0.5f / (float)BROWS);
}

// ---------------------------------------------------------------------------
// Host-side launcher
// ---------------------------------------------------------------------------
extern "C" void kernel_launch(void* const* d_in, const int* in_sizes, int n_in,
                              void* d_out, int out_size, void* d_ws, size_t ws_size,
                              hipStream_t stream)
{
    const float* pred  = (const float*)d_in[0];   // prediction (B, N) fp32
    const float* label = (const float*)d_in[1];   // label      (B, N) fp32
    float* out = (float*)d_out;

    // workspace layout: 4 int arrays [BROWS] + 1 float array [BROWS]
    int*   fr = (int*)d_ws;
    int*   lr = fr + BROWS;
    int*   fi = lr + BROWS;
    int*   li = fi + BROWS;
    float* row_mean = (float*)(li + BROWS);

    bounds_kernel<<<BROWS, 256, 0, stream>>>(label, fr, lr, fi, li);
    loss_kernel  <<<BROWS, 256, 0, stream>>>(pred, label, fr, lr, fi, li, row_mean);
    final_kernel <<<1, 32, 0, stream>>>(row_mean, out);
}